// Fume3dLayer_56418690401064
// MI455X (gfx1250) — compile-verified
//
#include <hip/hip_runtime.h>

// FUME epipolar translation for MI455X (gfx1250, wave32).
//
// Roofline: ~2 MB total HBM traffic, ~33M 2-tap samples -> gather-bound.
// Strategy: stage each 64 KB channel image into LDS (CDNA5's 320 KB/WGP
// budget makes this trivial) via async global->LDS copies (ASYNCcnt path),
// with a +1-float row pad so column-direction line walks are LDS
// bank-conflict-free, then do a branchless unified line walk per pixel.

#define AS1 __attribute__((address_space(1)))
#define AS3 __attribute__((address_space(3)))

namespace {
constexpr int IMG_H    = 128;
constexpr int IMG_W    = 128;
constexpr int LSTRIDE  = IMG_W + 1;  // 129: bank = (row + col) % 64 -> conflict-free column walks
constexpr int NTHREADS = 256;        // 8 waves (wave32)
constexpr int STAGE_ITERS = (IMG_H * IMG_W) / NTHREADS;  // 64, compile-time trip count
}

__global__ __launch_bounds__(NTHREADS)
void fume_epipolar_kernel(const float* __restrict__ img,
                          const float* __restrict__ Fm,
                          const float* __restrict__ dfac,
                          float* __restrict__ out,
                          int C) {
    __shared__ float tile[IMG_H * LSTRIDE];   // 66,048 B of the 320 KB WGP LDS

    const int tid = threadIdx.x;
    const int blk = blockIdx.x;   // 64 tiles of 256 pixels = 16384 pixels
    const int c   = blockIdx.y;
    const int b   = blockIdx.z;

    const float* __restrict__ src = img + (size_t)(b * C + c) * (IMG_H * IMG_W);

    // ---- Stage one channel image global -> LDS (row-padded layout) ----
    // Constant trip count -> straight-line fire-and-forget async issues.
    {
        const int row0 = tid >> 7;            // 0 or 1
        const int col  = tid & (IMG_W - 1);   // 0..127
#if defined(__has_builtin) && __has_builtin(__builtin_amdgcn_global_load_async_to_lds_b32)
#pragma unroll
        for (int it = 0; it < STAGE_ITERS; ++it) {
            const int row = row0 + it * (NTHREADS / IMG_W);   // +2 per iteration
            __builtin_amdgcn_global_load_async_to_lds_b32(
                (AS1 int*)(src + it * NTHREADS + tid),
                (AS3 int*)(&tile[row * LSTRIDE + col]),
                /*offset=*/0, /*cpol=*/0);
        }
#if __has_builtin(__builtin_amdgcn_s_wait_asynccnt)
        __builtin_amdgcn_s_wait_asynccnt(0);
#else
        asm volatile("s_wait_asynccnt 0" ::: "memory");
#endif
#else
#pragma unroll
        for (int it = 0; it < STAGE_ITERS; ++it) {
            const int row = row0 + it * (NTHREADS / IMG_W);
            tile[row * LSTRIDE + col] = src[it * NTHREADS + tid];
        }
#endif
    }
    __syncthreads();

    // ---- Per-pixel line coefficients (l = F @ [x*d, y*d, 1]) ----
    const float d   = dfac[b];
    const float f00 = Fm[0], f01 = Fm[1], f02 = Fm[2];
    const float f10 = Fm[3], f11 = Fm[4], f12 = Fm[5];
    const float f20 = Fm[6], f21 = Fm[7], f22 = Fm[8];

    const int p  = blk * NTHREADS + tid;     // 0..16383
    const int y2 = p >> 7;
    const int x2 = p & (IMG_W - 1);

    const float px = (float)x2 * d;
    const float py = (float)y2 * d;
    const float Ac = f00 * px + f01 * py + f02;
    const float Bc = f10 * px + f11 * py + f12;
    const float Cc = f20 * px + f21 * py + f22;

    const bool horiz = fabsf(Bc) >= fabsf(Ac);

    // Branchless unification of horizontal / vertical traversals (H == W):
    //   horiz:   step k over columns, q = -(A*(k*d)+C)/(B*d), sample (row=q, col=k)
    //   vert:    step k over rows,    q = -(B*(k*d)+C)/(A*d), sample (row=k, col=q)
    const float num = horiz ? Ac : Bc;
    const float den = horiz ? Bc : Ac;
    const float inv = 1.0f / (den * d);
    const int   mq  = horiz ? LSTRIDE : 1;   // multiplier for the interpolated coord
    const int   mk  = horiz ? 1 : LSTRIDE;   // multiplier for the stepping coord

    float acc = 0.0f;
#pragma unroll 4
    for (int k = 0; k < IMG_W; ++k) {
        const float q   = -fmaf(num, (float)k * d, Cc) * inv;
        const float q0f = floorf(q);
        const float wq  = q - q0f;
        int q0 = (int)q0f;
        q0 = q0 < 0 ? 0 : (q0 > IMG_H - 1 ? IMG_H - 1 : q0);
        const int q1 = (q0 + 1 > IMG_H - 1) ? (IMG_H - 1) : (q0 + 1);

        const int koff = k * mk;
        const float v0 = tile[q0 * mq + koff];
        const float v1 = tile[q1 * mq + koff];
        const float v  = (1.0f - wq) * v0 + wq * v1;

        const bool inb = (q >= 0.0f) && (q <= (float)(IMG_H - 1));
        acc += inb ? v : 0.0f;
    }

    out[((size_t)(b * C + c) * IMG_H + y2) * IMG_W + x2] = acc;
}

extern "C" void kernel_launch(void* const* d_in, const int* in_sizes, int n_in,
                              void* d_out, int out_size, void* d_ws, size_t ws_size,
                              hipStream_t stream) {
    const float* view1 = (const float*)d_in[0];   // (B, C, 128, 128) f32
    const float* F21   = (const float*)d_in[1];   // (3, 3) f32
    // d_in[2] = F12 (unused in forward)
    const float* dfac  = (const float*)d_in[3];   // (B,) f32

    const int B = in_sizes[3];
    const int C = in_sizes[0] / (B * IMG_H * IMG_W);

    dim3 grid((IMG_H * IMG_W) / NTHREADS, C, B);  // 64 x C x B
    dim3 block(NTHREADS);
    fume_epipolar_kernel<<<grid, block, 0, stream>>>(view1, F21, dfac, (float*)d_out, C);
}